// GR_NCF_20091857010782
// MI455X (gfx1250) — compile-verified
//
#include <hip/hip_runtime.h>
#include <hip/hip_bf16.h>

typedef float v2f __attribute__((ext_vector_type(2)));
typedef float v8f __attribute__((ext_vector_type(8)));

#define EMBED 32
#define HID   48
#define WAVES 4
#define SA 100   // bufA stride (ncf width 96 + pad, even for b64-aligned ds loads)
#define SB 52    // bufB stride (width 48 + pad, even)

__device__ __forceinline__ v8f wmma4(v2f a, v2f b, v8f c) {
  // D = A(16x4 f32) * B(4x16 f32) + C(16x16 f32)
  return __builtin_amdgcn_wmma_f32_16x16x4_f32(false, a, false, b, (short)0, c,
                                               false, false);
}

// ---------------------------------------------------------------------------
// Kernel 1: segment mean + relu.  One wave per group; lane = embedding dim.
// segment_ids are sorted, so each wave binary-searches its [start,end) range
// and streams 128B user_table rows fully coalesced.
// ---------------------------------------------------------------------------
__global__ __launch_bounds__(256) void seg_mean_relu(
    const int* __restrict__ member_ids, const int* __restrict__ segment_ids,
    const float* __restrict__ user_table, float* __restrict__ agg,
    int T, int B) {
  const int lane = threadIdx.x & 31;
  const int wave = threadIdx.x >> 5;
  const int g = blockIdx.x * (blockDim.x >> 5) + wave;
  if (g >= B) return;

  int lo = 0, hi = T;
  while (lo < hi) { int mid = (lo + hi) >> 1;
    if (segment_ids[mid] < g) lo = mid + 1; else hi = mid; }
  const int start = lo;
  hi = T;
  while (lo < hi) { int mid = (lo + hi) >> 1;
    if (segment_ids[mid] < g + 1) lo = mid + 1; else hi = mid; }
  const int end = lo;

  float s = 0.f;
  for (int t = start; t < end; ++t) {
    int uid = member_ids[t];
    s += user_table[(size_t)uid * EMBED + lane];
  }
  int cnt = end - start;
  float v = s / (float)(cnt > 0 ? cnt : 1);
  agg[(size_t)g * EMBED + lane] = v > 0.f ? v : 0.f;
}

// ---------------------------------------------------------------------------
// Kernel 2: fully fused MLP via V_WMMA_F32_16X16X4_F32.
// One wave = one 16-row tile.  A-frag: lane(m=lane&15,h=lane>>4) holds
// {A[m][4c+2h], A[m][4c+2h+1]}; B-frag: {W[4c+2h][n], W[4c+2h+1][n]} with
// n = nt*16 + m; D: acc[r] = D[8h+r][m].
// ---------------------------------------------------------------------------
__global__ __launch_bounds__(WAVES * 32) void fused_mlp(
    const float* __restrict__ agg, const int* __restrict__ item_inputs,
    const float* __restrict__ item_table,
    const float* __restrict__ w1, const float* __restrict__ b1,
    const float* __restrict__ w2, const float* __restrict__ b2,
    const float* __restrict__ w3, const float* __restrict__ b3,
    const float* __restrict__ wp1, const float* __restrict__ bp1,
    const float* __restrict__ wp2, const float* __restrict__ bp2,
    float* __restrict__ out) {
  __shared__ __align__(16) float bufA_s[WAVES][16 * SA];
  __shared__ __align__(16) float bufB_s[WAVES][16 * SB];
  const int lane = threadIdx.x & 31;
  const int wave = threadIdx.x >> 5;
  const int g0 = (blockIdx.x * WAVES + wave) * 16;
  const int m = lane & 15;
  const int h = lane >> 4;
  float* A  = bufA_s[wave];
  float* Bb = bufB_s[wave];

  // ---- Layer 1: agg[16x32] @ w1[32x48] + b1, relu  -> Bb (h1, cols 0..47)
  v2f a1[8];
  const float* arow = agg + (size_t)(g0 + m) * EMBED;
#pragma unroll
  for (int c = 0; c < 8; ++c) a1[c] = *(const v2f*)(arow + 4 * c + 2 * h);

  for (int nt = 0; nt < 3; ++nt) {
    const int n = nt * 16 + m;
    const float bias = b1[n];
    v8f acc;
#pragma unroll
    for (int r = 0; r < 8; ++r) acc[r] = bias;
    for (int c = 0; c < 8; ++c) {
      v2f bf;
      bf.x = w1[(4 * c + 2 * h) * HID + n];
      bf.y = w1[(4 * c + 2 * h + 1) * HID + n];
      acc = wmma4(a1[c], bf, acc);
    }
#pragma unroll
    for (int r = 0; r < 8; ++r) {
      float v = acc[r];
      Bb[(8 * h + r) * SB + n] = v > 0.f ? v : 0.f;
    }
  }
  asm volatile("s_wait_dscnt 0" ::: "memory");

  // ---- Layer 2: h1[16x48] @ w2[48x48] + b2, relu  -> A (h2, cols 0..47)
  for (int nt = 0; nt < 3; ++nt) {
    const int n = nt * 16 + m;
    const float bias = b2[n];
    v8f acc;
#pragma unroll
    for (int r = 0; r < 8; ++r) acc[r] = bias;
    for (int c = 0; c < 12; ++c) {
      v2f af = *(const v2f*)(Bb + m * SB + 4 * c + 2 * h);
      v2f bf;
      bf.x = w2[(4 * c + 2 * h) * HID + n];
      bf.y = w2[(4 * c + 2 * h + 1) * HID + n];
      acc = wmma4(af, bf, acc);
    }
#pragma unroll
    for (int r = 0; r < 8; ++r) {
      float v = acc[r];
      A[(8 * h + r) * SA + n] = v > 0.f ? v : 0.f;
    }
  }
  asm volatile("s_wait_dscnt 0" ::: "memory");

  // ---- Layer 3 (z half only): h2[16x48] @ w3[48,0:32] + b3 -> Bb cols 0..31
  for (int nt = 0; nt < 2; ++nt) {
    const int n = nt * 16 + m;
    const float bias = b3[n];
    v8f acc;
#pragma unroll
    for (int r = 0; r < 8; ++r) acc[r] = bias;
    for (int c = 0; c < 12; ++c) {
      v2f af = *(const v2f*)(A + m * SA + 4 * c + 2 * h);
      v2f bf;
      bf.x = w3[(4 * c + 2 * h) * (2 * EMBED) + n];
      bf.y = w3[(4 * c + 2 * h + 1) * (2 * EMBED) + n];
      acc = wmma4(af, bf, acc);
    }
#pragma unroll
    for (int r = 0; r < 8; ++r) Bb[(8 * h + r) * SB + n] = acc[r];  // z, no relu
  }
  asm volatile("s_wait_dscnt 0" ::: "memory");

  // ---- Item gather -> A cols 64..95 (h2 is dead now)
  for (int i = lane; i < 16 * EMBED; i += 32) {
    int r = i >> 5, cI = i & 31;
    int iid = item_inputs[g0 + r];
    A[r * SA + 64 + cI] = item_table[(size_t)iid * EMBED + cI];
  }
  asm volatile("s_wait_dscnt 0" ::: "memory");

  // ---- ncf = [z*item | z | item] in A cols 0..95
  for (int i = lane; i < 16 * EMBED; i += 32) {
    int r = i >> 5, cI = i & 31;
    float z  = Bb[r * SB + cI];
    float it = A[r * SA + 64 + cI];
    A[r * SA + 32 + cI] = z;
    A[r * SA + cI] = z * it;
  }
  asm volatile("s_wait_dscnt 0" ::: "memory");

  // ---- Predict: ncf[16x96] @ wp1[96x8] (N padded to 16) + bp1, relu,
  //      then fold wp2 per-column and reduce.
  {
    const int n = m;
    const float bias = (n < 8) ? bp1[n] : 0.f;
    v8f acc;
#pragma unroll
    for (int r = 0; r < 8; ++r) acc[r] = bias;
    for (int c = 0; c < 24; ++c) {
      v2f af = *(const v2f*)(A + m * SA + 4 * c + 2 * h);
      v2f bf;
      bf.x = (n < 8) ? wp1[(4 * c + 2 * h) * 8 + n] : 0.f;
      bf.y = (n < 8) ? wp1[(4 * c + 2 * h + 1) * 8 + n] : 0.f;
      acc = wmma4(af, bf, acc);
    }
    const float wn = (n < 8) ? wp2[n] : 0.f;
#pragma unroll
    for (int r = 0; r < 8; ++r) {
      float v = acc[r];
      v = v > 0.f ? v : 0.f;
      Bb[(8 * h + r) * SB + n] = v * wn;   // cols 8..15 are zeros
    }
  }
  asm volatile("s_wait_dscnt 0" ::: "memory");

  // ---- Row reduce + sigmoid (lanes 0..15, after all WMMAs)
  if (lane < 16) {
    float s = 0.f;
#pragma unroll
    for (int j = 0; j < 8; ++j) s += Bb[lane * SB + j];
    s += bp2[0];
    out[g0 + lane] = 1.f / (1.f + __expf(-s));
  }
}

// ---------------------------------------------------------------------------
extern "C" void kernel_launch(void* const* d_in, const int* in_sizes, int n_in,
                              void* d_out, int out_size, void* d_ws,
                              size_t ws_size, hipStream_t stream) {
  (void)n_in; (void)out_size; (void)ws_size;
  // setup_inputs order:
  // 0 group_inputs (unused by reference), 1 item_inputs, 2 member_ids,
  // 3 segment_ids, 4 user_table, 5 item_table, 6..15 weights/biases
  const int*   item_inputs = (const int*)d_in[1];
  const int*   member_ids  = (const int*)d_in[2];
  const int*   segment_ids = (const int*)d_in[3];
  const float* user_table  = (const float*)d_in[4];
  const float* item_table  = (const float*)d_in[5];
  const float* w1  = (const float*)d_in[6];
  const float* b1  = (const float*)d_in[7];
  const float* w2  = (const float*)d_in[8];
  const float* b2  = (const float*)d_in[9];
  const float* w3  = (const float*)d_in[10];
  const float* b3  = (const float*)d_in[11];
  const float* wp1 = (const float*)d_in[12];
  const float* bp1 = (const float*)d_in[13];
  const float* wp2 = (const float*)d_in[14];
  const float* bp2 = (const float*)d_in[15];

  const int B = in_sizes[1];   // 16384
  const int T = in_sizes[2];   // 327680

  float* agg = (float*)d_ws;   // [B, 32] scratch

  seg_mean_relu<<<(B + 7) / 8, 256, 0, stream>>>(member_ids, segment_ids,
                                                 user_table, agg, T, B);

  const int tiles = B / 16;                 // 1024
  fused_mlp<<<tiles / WAVES, WAVES * 32, 0, stream>>>(
      agg, item_inputs, item_table, w1, b1, w2, b2, w3, b3, wp1, bp1, wp2, bp2,
      (float*)d_out);
}